// LocalContinuityLoss_40226663694453
// MI455X (gfx1250) — compile-verified
//
#include <hip/hip_runtime.h>
#include <math.h>

// ---------------------------------------------------------------------------
// LocalContinuityLoss for MI455X (gfx1250, wave32, WMMA)
//
// Phase 1 (x2, pred/target): fused Gram-matrix (V_WMMA_F32_16X16X4_F32,
//   coords padded to K=4) + fast-sqrt distance + per-row top-8 selection.
//   One wave per 16-row panel; WMMA C layout transposed via LDS tile.
// Phase 2: per-point density/direction/covariance contributions, block-level
//   deterministic LDS reduction into 64 partials.
// Phase 3: finalize scalar.
// ---------------------------------------------------------------------------

typedef __attribute__((ext_vector_type(2))) float v2f;
typedef __attribute__((ext_vector_type(8))) float v8f;

#define NB    4
#define NPTS  4096
#define KNB   8
#define EPSF  1e-12f
#define BIGF  3.0e38f
#define NTILES (NPTS / 16)
#define WAVES_PER_BLOCK 4

struct WaveLds {
    float sqA[16];
    float dist[16][16];
};

__launch_bounds__(WAVES_PER_BLOCK * 32)
__global__ void knn_wmma_kernel(const float* __restrict__ pts,
                                float* __restrict__ density,
                                int*   __restrict__ nbr)
{
    __shared__ WaveLds lds[WAVES_PER_BLOCK];

    const int lane  = threadIdx.x & 31;
    const int wv    = threadIdx.x >> 5;
    const int panel = blockIdx.x * WAVES_PER_BLOCK + wv;   // 0 .. NB*NTILES-1
    const int b       = panel / NTILES;
    const int rowBase = (panel % NTILES) << 4;
    const float* __restrict__ P = pts + (size_t)b * NPTS * 3;

    const int lm = lane & 15;   // column / row-in-tile index
    const int hi = lane >> 4;   // K-half selector (K0,K1 vs K2,K3)

    // ---- A tile: 16 rows x 4 K (z padded with 0 in K3) -------------------
    const int arow = rowBase + lm;
    const float ax = P[arow * 3 + 0];
    const float ay = P[arow * 3 + 1];
    const float az = P[arow * 3 + 2];
    v2f A;
    A.x = hi ? az   : ax;
    A.y = hi ? 0.0f : ay;

    if (lane < 16) lds[wv].sqA[lm] = ax * ax + ay * ay + az * az;
    asm volatile("s_wait_dscnt 0" ::: "memory");

    float sqRow[8];
#pragma unroll
    for (int v = 0; v < 8; ++v) sqRow[v] = lds[wv].sqA[v + (hi << 3)];

    // ---- per-row top-8 state (lanes 0-15 own one row each) ---------------
    float bd[KNB];
    int   bi[KNB];
#pragma unroll
    for (int k = 0; k < KNB; ++k) { bd[k] = BIGF; bi[k] = 0; }

    // ---- sweep all 4096 columns in 16-wide tiles -------------------------
    for (int ct = 0; ct < NTILES; ++ct) {
        const int col = (ct << 4) + lm;
        const float bx = P[col * 3 + 0];
        const float by = P[col * 3 + 1];
        const float bz = P[col * 3 + 2];
        if (ct + 1 < NTILES) __builtin_prefetch(&P[(col + 16) * 3], 0, 1);

        v2f Bv;
        Bv.x = hi ? bz   : bx;
        Bv.y = hi ? 0.0f : by;
        const float sqb = bx * bx + by * by + bz * bz;

        v8f c = {};
        // D = A(16x4) * B(4x16) + 0 : one 16x16 Gram tile per instruction
        c = __builtin_amdgcn_wmma_f32_16x16x4_f32(
                /*neg_a=*/false, A, /*neg_b=*/false, Bv,
                /*c_mod=*/(short)0, c, /*reuse_a=*/false, /*reuse_b=*/false);

        // d2 = |a|^2 + |b|^2 - 2 a.b ; transpose to row-owner lanes via LDS
#pragma unroll
        for (int v = 0; v < 8; ++v) {
            const int m  = v + (hi << 3);
            const float d2 = sqRow[v] + sqb - 2.0f * c[v];
            // single-instruction v_sqrt_f32: ordering-preserving, ~1 ulp
            float d = __builtin_amdgcn_sqrtf(fmaxf(d2, EPSF));
            if (rowBase + m == col) d = BIGF;   // exclude self
            lds[wv].dist[m][lm] = d;
        }
        asm volatile("s_wait_dscnt 0" ::: "memory");

        if (lane < 16) {
#pragma unroll 4
            for (int j = 0; j < 16; ++j) {
                const float d = lds[wv].dist[lane][j];
                if (d < bd[KNB - 1]) {
                    const int cidx = (ct << 4) + j;
                    int pos = KNB - 1;
#pragma unroll
                    for (int k = KNB - 2; k >= 0; --k) {
                        if (d < bd[k]) { bd[k + 1] = bd[k]; bi[k + 1] = bi[k]; pos = k; }
                    }
                    bd[pos] = d; bi[pos] = cidx;
                }
            }
        }
        // DS ops are in-order per wave: next iteration's ds_store cannot
        // bypass this iteration's ds_loads, no extra wait needed here.
    }

    if (lane < 16) {
        const int row = rowBase + lane;
        float s = 0.0f;
#pragma unroll
        for (int k = 0; k < KNB; ++k) s += bd[k];
        density[b * NPTS + row] = s * (1.0f / KNB);
#pragma unroll
        for (int k = 0; k < KNB; ++k)
            nbr[((size_t)b * NPTS + row) * KNB + k] = bi[k];
    }
}

__launch_bounds__(256)
__global__ void point_loss_kernel(const float* __restrict__ pred,
                                  const float* __restrict__ target,
                                  const float* __restrict__ densP,
                                  const float* __restrict__ densT,
                                  const int*   __restrict__ idxP,
                                  const int*   __restrict__ idxT,
                                  float* __restrict__ partials)
{
    const int i = blockIdx.x * blockDim.x + threadIdx.x;   // 0 .. NB*NPTS-1
    const int b = i / NPTS;
    const int n = i - b * NPTS;
    const float* __restrict__ Pp = pred   + (size_t)b * NPTS * 3;
    const float* __restrict__ Pt = target + (size_t)b * NPTS * 3;

    const float p0x = Pp[n * 3 + 0], p0y = Pp[n * 3 + 1], p0z = Pp[n * 3 + 2];
    const float t0x = Pt[n * 3 + 0], t0y = Pt[n * 3 + 1], t0z = Pt[n * 3 + 2];

    const float dd = densP[i] - densT[i];
    float contrib = dd * dd;

    float cpxx = 0, cpyy = 0, cpzz = 0, cpxy = 0, cpxz = 0, cpyz = 0;
    float ctxx = 0, ctyy = 0, ctzz = 0, ctxy = 0, ctxz = 0, ctyz = 0;
    float dots = 0;

#pragma unroll
    for (int k = 0; k < KNB; ++k) {
        const int jp = idxP[(size_t)i * KNB + k];
        const int jt = idxT[(size_t)i * KNB + k];
        const float pvx = Pp[jp * 3 + 0] - p0x;
        const float pvy = Pp[jp * 3 + 1] - p0y;
        const float pvz = Pp[jp * 3 + 2] - p0z;
        const float tvx = Pt[jt * 3 + 0] - t0x;
        const float tvy = Pt[jt * 3 + 1] - t0y;
        const float tvz = Pt[jt * 3 + 2] - t0z;
        const float np = fmaxf(__builtin_amdgcn_sqrtf(pvx * pvx + pvy * pvy + pvz * pvz), EPSF);
        const float nt = fmaxf(__builtin_amdgcn_sqrtf(tvx * tvx + tvy * tvy + tvz * tvz), EPSF);
        dots += (pvx * tvx + pvy * tvy + pvz * tvz) * __builtin_amdgcn_rcpf(np * nt);
        cpxx += pvx * pvx; cpyy += pvy * pvy; cpzz += pvz * pvz;
        cpxy += pvx * pvy; cpxz += pvx * pvz; cpyz += pvy * pvz;
        ctxx += tvx * tvx; ctyy += tvy * tvy; ctzz += tvz * tvz;
        ctxy += tvx * tvy; ctxz += tvx * tvz; ctyz += tvy * tvz;
    }

    const float invK = 1.0f / KNB;
    const float dxx = (cpxx - ctxx) * invK;
    const float dyy = (cpyy - ctyy) * invK;
    const float dzz = (cpzz - ctzz) * invK;
    const float dxy = (cpxy - ctxy) * invK;
    const float dxz = (cpxz - ctxz) * invK;
    const float dyz = (cpyz - ctyz) * invK;
    const float fro = __builtin_amdgcn_sqrtf(dxx * dxx + dyy * dyy + dzz * dzz +
                                             2.0f * (dxy * dxy + dxz * dxz + dyz * dyz));

    contrib += 0.5f * fro - 0.5f * dots * invK;

    // deterministic block reduction
    __shared__ float red[256];
    red[threadIdx.x] = contrib;
    __syncthreads();
#pragma unroll
    for (int s = 128; s > 0; s >>= 1) {
        if (threadIdx.x < s) red[threadIdx.x] += red[threadIdx.x + s];
        __syncthreads();
    }
    if (threadIdx.x == 0) partials[blockIdx.x] = red[0];
}

__global__ void finalize_kernel(const float* __restrict__ partials,
                                float* __restrict__ out, int nparts)
{
    float s = 0.0f;
    for (int k = 0; k < nparts; ++k) s += partials[k];
    out[0] = 0.5f + s * (1.0f / (float)(NB * NPTS));
}

extern "C" void kernel_launch(void* const* d_in, const int* in_sizes, int n_in,
                              void* d_out, int out_size, void* d_ws, size_t ws_size,
                              hipStream_t stream)
{
    (void)in_sizes; (void)n_in; (void)out_size; (void)ws_size;
    const float* pred   = (const float*)d_in[0];
    const float* target = (const float*)d_in[1];
    float* out = (float*)d_out;

    char* ws = (char*)d_ws;
    const size_t BN = (size_t)NB * NPTS;
    float* densP = (float*)ws;
    float* densT = densP + BN;
    int*   idxP  = (int*)(densT + BN);
    int*   idxT  = idxP + BN * KNB;
    float* parts = (float*)(idxT + BN * KNB);

    const int knnBlocks = (NB * NTILES) / WAVES_PER_BLOCK;      // 256
    knn_wmma_kernel<<<knnBlocks, WAVES_PER_BLOCK * 32, 0, stream>>>(pred,   densP, idxP);
    knn_wmma_kernel<<<knnBlocks, WAVES_PER_BLOCK * 32, 0, stream>>>(target, densT, idxT);

    const int ptBlocks = (int)(BN / 256);                       // 64
    point_loss_kernel<<<ptBlocks, 256, 0, stream>>>(pred, target, densP, densT,
                                                    idxP, idxT, parts);
    finalize_kernel<<<1, 1, 0, stream>>>(parts, out, ptBlocks);
}